// GCRNN_41729902248421
// MI455X (gfx1250) — compile-verified
//
#include <hip/hip_runtime.h>
#include <hip/hip_bf16.h>

#define U_CNT 100000
#define N_CNT 50000
#define C_CNT 20
#define D_DIM 128
#define E_CNT 1000000
#define K_CNT 50000
#define NS_CNT 9
#define G4D 512
#define ASTRIDE 260  // padded LDS pitch for A tile (avoids 16-way bank conflicts)

typedef float v2f __attribute__((ext_vector_type(2)));
typedef float v8f __attribute__((ext_vector_type(8)));

// ---------------- zero scratch (agg + cnt) ----------------
__global__ void zero_ws_kernel(float* __restrict__ p, int n) {
    int i = blockIdx.x * blockDim.x + threadIdx.x;
    int stride = gridDim.x * blockDim.x;
    for (; i < n; i += stride) p[i] = 0.0f;
}

// ---------------- pack B = [W_ih ; W_hh]^T into WMMA B-fragment order ----
// B[k][n], k in [0,256), n in [0,512).  Fragment layout guess per ISA A/C
// conventions: lanes 0-15 hold K = k0+{0,1}, lanes 16-31 hold K = k0+{2,3},
// N = ntBase + lane%16.  Linear index t = ((nt*64 + ks)*64 + lane*2 + j).
__global__ void pack_b_kernel(const float* __restrict__ Wih,
                              const float* __restrict__ Whh,
                              float* __restrict__ Bp) {
    int t = blockIdx.x * 256 + threadIdx.x;   // 131072 total elements
    int j    = t & 1;
    int lane = (t >> 1) & 31;
    int ks   = (t >> 6) & 63;
    int nt   = t >> 12;
    int n = nt * 16 + (lane & 15);
    int k = ks * 4 + ((lane < 16) ? j : 2 + j);
    float v = (k < 128) ? Wih[n * 128 + k] : Whh[n * 128 + (k - 128)];
    Bp[t] = v;
}

// ---------------- GCN edge scatter: agg += news[src]*cat[ecat], cnt += 1 ---
__global__ void edge_scatter_kernel(const float* __restrict__ news,
                                    const float* __restrict__ cat,
                                    const int* __restrict__ esrc,
                                    const int* __restrict__ edst,
                                    const int* __restrict__ ecat,
                                    float* __restrict__ agg,
                                    float* __restrict__ cnt) {
    __shared__ __align__(16) float scat[C_CNT * D_DIM];  // 10 KB
    for (int i = threadIdx.x; i < C_CNT * D_DIM; i += 256) scat[i] = cat[i];
    __syncthreads();

    int wv = blockIdx.x * 8 + (threadIdx.x >> 5);
    if (wv >= E_CNT) return;
    int lane = threadIdx.x & 31;
    int src = esrc[wv], dst = edst[wv], ct = ecat[wv];

    const float4 nv = *(const float4*)(news + (size_t)src * 128 + lane * 4);
    const float4 cv = *(const float4*)(scat + ct * 128 + lane * 4);
    float* a = agg + (size_t)dst * 128 + lane * 4;
    atomicAdd(a + 0, nv.x * cv.x);
    atomicAdd(a + 1, nv.y * cv.y);
    atomicAdd(a + 2, nv.z * cv.z);
    atomicAdd(a + 3, nv.w * cv.w);
    if (lane == 0) atomicAdd(cnt + dst, 1.0f);
}

// ---------------- fused LSTM cell via f32 WMMA ----------------------------
// gates = [x | h] (32x256 tile) @ Bpack (256x512), + biases, then pointwise.
__global__ __launch_bounds__(256) void lstm_wmma_kernel(
        const float* __restrict__ uemb, const float* __restrict__ cx,
        const float* __restrict__ bih,  const float* __restrict__ bhh,
        const float* __restrict__ agg,  const float* __restrict__ cnt,
        const float* __restrict__ Bp,   float* __restrict__ hnew) {
    __shared__ float smem[32 * 512];   // 64 KB: A tile first, gates after
    const int tid = threadIdx.x;
    const int ubase = blockIdx.x * 32;

    // Stage A tile: A[u][k] = (k<128) ? x[u][k] : user_emb[u][k-128]
    for (int e = tid; e < 32 * 256; e += 256) {
        int u = e >> 8, k = e & 255;
        int gu = ubase + u;
        float v;
        if (k < 128) {
            float c = cnt[gu];
            v = uemb[(size_t)gu * 128 + k] +
                agg[(size_t)gu * 128 + k] / fmaxf(c, 1.0f);
        } else {
            v = uemb[(size_t)gu * 128 + (k - 128)];
        }
        smem[u * ASTRIDE + k] = v;
    }
    __syncthreads();

    const int wave = tid >> 5;
    const int lane = tid & 31;
    const int halfsel = (lane < 16) ? 0 : 2;   // A/B K split across half-waves
    const int mrow = lane & 15;

    v8f acc[2][4];
    const v8f vzero = {0.f, 0.f, 0.f, 0.f, 0.f, 0.f, 0.f, 0.f};
#pragma unroll
    for (int m = 0; m < 2; ++m)
#pragma unroll
        for (int t = 0; t < 4; ++t) acc[m][t] = vzero;

    const float* bpw = Bp + (size_t)(wave * 4) * 4096;  // this wave's 4 N-tiles
    for (int ks = 0; ks < 64; ++ks) {
        const int k0 = ks * 4;
        v2f a0, a1;
        a0.x = smem[mrow * ASTRIDE + k0 + halfsel];
        a0.y = smem[mrow * ASTRIDE + k0 + halfsel + 1];
        a1.x = smem[(16 + mrow) * ASTRIDE + k0 + halfsel];
        a1.y = smem[(16 + mrow) * ASTRIDE + k0 + halfsel + 1];
#pragma unroll
        for (int t = 0; t < 4; ++t) {
            v2f b = *(const v2f*)(bpw + t * 4096 + ks * 64 + lane * 2);
            acc[0][t] = __builtin_amdgcn_wmma_f32_16x16x4_f32(
                false, a0, false, b, (short)0, acc[0][t], false, false);
            acc[1][t] = __builtin_amdgcn_wmma_f32_16x16x4_f32(
                false, a1, false, b, (short)0, acc[1][t], false, false);
        }
    }
    __syncthreads();

    // Spill C fragments -> gates[32][512] in LDS (ISA C layout: VGPR r holds
    // rows r (lanes 0-15) and r+8 (lanes 16-31), N = lane%16).
#pragma unroll
    for (int m = 0; m < 2; ++m)
#pragma unroll
        for (int t = 0; t < 4; ++t)
#pragma unroll
            for (int r = 0; r < 8; ++r) {
                int row = m * 16 + ((lane < 16) ? r : r + 8);
                int col = (wave * 4 + t) * 16 + (lane & 15);
                smem[row * 512 + col] = acc[m][t][r];
            }
    __syncthreads();

    // Pointwise LSTM epilogue: i,f,g,o -> h_new
    for (int e = tid; e < 32 * 128; e += 256) {
        int u = e >> 7, d = e & 127;
        int gu = ubase + u;
        float ig = smem[u * 512 + d]       + bih[d]       + bhh[d];
        float fg = smem[u * 512 + 128 + d] + bih[128 + d] + bhh[128 + d];
        float gg = smem[u * 512 + 256 + d] + bih[256 + d] + bhh[256 + d];
        float og = smem[u * 512 + 384 + d] + bih[384 + d] + bhh[384 + d];
        float si = 1.0f / (1.0f + __expf(-ig));
        float sf = 1.0f / (1.0f + __expf(-fg));
        float so = 1.0f / (1.0f + __expf(-og));
        float c  = sf * cx[(size_t)gu * 128 + d] + si * tanhf(gg);
        hnew[(size_t)gu * 128 + d] = so * tanhf(c);
    }
}

// ---------------- candidate scoring: wave per query, shuffle reduction ----
__global__ void score_kernel(const float* __restrict__ hnew,
                             const float* __restrict__ news,
                             const int* __restrict__ nsidx,
                             const int* __restrict__ usidx,
                             float* __restrict__ out) {
    int wv = blockIdx.x * 8 + (threadIdx.x >> 5);
    if (wv >= K_CNT) return;
    int lane = threadIdx.x & 31;
    int uid = usidx[wv];
    const float4 cu = *(const float4*)(hnew + (size_t)uid * 128 + lane * 4);
#pragma unroll
    for (int n = 0; n < NS_CNT; ++n) {
        int nid = nsidx[wv * NS_CNT + n];
        const float4 cn = *(const float4*)(news + (size_t)nid * 128 + lane * 4);
        float p = cu.x * cn.x + cu.y * cn.y + cu.z * cn.z + cu.w * cn.w;
        for (int off = 16; off > 0; off >>= 1) p += __shfl_down(p, off, 32);
        if (lane == 0) out[wv * NS_CNT + n] = p;
    }
}

extern "C" void kernel_launch(void* const* d_in, const int* in_sizes, int n_in,
                              void* d_out, int out_size, void* d_ws, size_t ws_size,
                              hipStream_t stream) {
    const float* user_emb = (const float*)d_in[0];
    const float* news_emb = (const float*)d_in[1];
    const float* cat_emb  = (const float*)d_in[2];
    const float* cx       = (const float*)d_in[3];
    const float* W_ih     = (const float*)d_in[4];
    const float* W_hh     = (const float*)d_in[5];
    const float* b_ih     = (const float*)d_in[6];
    const float* b_hh     = (const float*)d_in[7];
    const int* edge_src   = (const int*)d_in[8];
    const int* edge_dst   = (const int*)d_in[9];
    const int* edge_cat   = (const int*)d_in[10];
    const int* ns_idx     = (const int*)d_in[11];
    const int* us_idx     = (const int*)d_in[12];

    float* ws   = (float*)d_ws;
    float* agg  = ws;                                   // U*D
    float* cnt  = agg  + (size_t)U_CNT * D_DIM;         // U
    float* hnew = cnt  + U_CNT;                         // U*D
    float* Bp   = hnew + (size_t)U_CNT * D_DIM;         // 256*512

    zero_ws_kernel<<<2048, 256, 0, stream>>>(agg, U_CNT * D_DIM + U_CNT);
    pack_b_kernel<<<(256 * 512 * 2) / 256 / 2, 256, 0, stream>>>(W_ih, W_hh, Bp); // 512 blocks
    edge_scatter_kernel<<<E_CNT / 8, 256, 0, stream>>>(news_emb, cat_emb,
                                                       edge_src, edge_dst, edge_cat,
                                                       agg, cnt);
    lstm_wmma_kernel<<<U_CNT / 32, 256, 0, stream>>>(user_emb, cx, b_ih, b_hh,
                                                     agg, cnt, Bp, hnew);
    score_kernel<<<K_CNT / 8, 256, 0, stream>>>(hnew, news_emb, ns_idx, us_idx,
                                                (float*)d_out);
}